// AttentionPageExample_45947560132807
// MI455X (gfx1250) — compile-verified
//
#include <hip/hip_runtime.h>
#include <stdint.h>

#ifndef __has_builtin
#define __has_builtin(x) 0
#endif

// ---- problem shape ----
#define NBS        4
#define NSEQ       128
#define NLAYERS    6
#define NHEADS     32
#define NHEAD_DIM  128
#define NSTRIDE    16
#define NFEAT      (NHEADS * NHEAD_DIM)          // 4096
#define NBLOCKS    (NSEQ / NSTRIDE)              // 8
#define NPAGES     128
#define NCACHE     2
#define PAGE_STR   (NLAYERS * NCACHE)            // 12
#define NSUB       (NPAGES * PAGE_STR)           // 1536 sub-blocks
#define SUB_ELEMS  (NSTRIDE * NHEADS * NHEAD_DIM) // 65536 floats per sub-block
#define SUB_V4     (SUB_ELEMS / 4)               // 16384 float4
#define H_ELEMS    (NBS * NSEQ * NFEAT)          // 2097152 floats

typedef float v4f __attribute__((ext_vector_type(4)));
typedef int   v4i __attribute__((ext_vector_type(4)));

// fp32 recurrence, exact op order of the reference (no FMA contraction):
// h=1; per layer: xk=2h, xv=4h, h = h + (h*xk)*xv
__device__ __forceinline__ void h_sequence(float hl[NLAYERS + 1]) {
#pragma clang fp contract(off)
  float h = 1.0f;
  for (int l = 0; l < NLAYERS; ++l) {
    hl[l] = h;
    float xk = 2.0f * h;
    float xv = 4.0f * h;
    h = h + (h * xk) * xv;
  }
  hl[NLAYERS] = h;
}

// ---- CDNA5 async global<->LDS data movement (ASYNCcnt path) ----
#if defined(__HIP_DEVICE_COMPILE__)
#if __has_builtin(__builtin_amdgcn_global_load_async_to_lds_b128) && \
    __has_builtin(__builtin_amdgcn_global_store_async_from_lds_b128)
#define ASYNC_BUILTIN 1
#endif
#endif

typedef __attribute__((address_space(1))) v4i* as1_v4i_p;
typedef __attribute__((address_space(3))) v4i* as3_v4i_p;

__device__ __forceinline__ void async_ld_b128(void* lds, const void* g) {
#if defined(ASYNC_BUILTIN)
  __builtin_amdgcn_global_load_async_to_lds_b128(
      (as1_v4i_p)(void*)g, (as3_v4i_p)lds, 0, 0);
#elif defined(__HIP_DEVICE_COMPILE__)
  unsigned loff = (unsigned)(uint64_t)(__attribute__((address_space(3))) char*)lds;
  asm volatile("global_load_async_to_lds_b128 %0, %1, off"
               :: "v"(loff), "v"(g) : "memory");
#else
  (void)lds; (void)g;
#endif
}

__device__ __forceinline__ void async_st_b128(void* g, const void* lds) {
#if defined(ASYNC_BUILTIN)
  __builtin_amdgcn_global_store_async_from_lds_b128(
      (as1_v4i_p)g, (as3_v4i_p)(void*)lds, 0, 0);
#elif defined(__HIP_DEVICE_COMPILE__)
  unsigned loff = (unsigned)(uint64_t)(__attribute__((address_space(3))) char*)(void*)lds;
  asm volatile("global_store_async_from_lds_b128 %0, %1, off"
               :: "v"(g), "v"(loff) : "memory");
#else
  (void)g; (void)lds;
#endif
}

__device__ __forceinline__ void wait_async0() {
#if defined(__HIP_DEVICE_COMPILE__)
#if __has_builtin(__builtin_amdgcn_s_wait_asynccnt)
  __builtin_amdgcn_s_wait_asynccnt(0);
#else
  asm volatile("s_wait_asynccnt 0" ::: "memory");
#endif
#endif
}

// ---- kernel 1: build per-sub-block mode table (in d_ws) ----
// table[sb] == -1           -> copy input sub-block
// table[sb] == float bits c -> fill sub-block with constant c
__global__ __launch_bounds__(512) void build_table(const int* __restrict__ ids,
                                                   int* __restrict__ table) {
  for (int i = threadIdx.x; i < NSUB; i += 512) table[i] = -1;
  __syncthreads();
  if (threadIdx.x < NBS * NBLOCKS) {
    float hl[NLAYERS + 1];
    h_sequence(hl);
    const int id = ids[threadIdx.x];
    for (int l = 0; l < NLAYERS; ++l) {
      const int kb = id * PAGE_STR + l * NCACHE;
      table[kb]     = __float_as_int(2.0f * hl[l]);  // K line
      table[kb + 1] = __float_as_int(4.0f * hl[l]);  // V line
    }
  }
}

// ---- kernel 2: h output is one constant; stream it with NT B128 stores ----
__global__ __launch_bounds__(256) void fill_h_kernel(v4f* __restrict__ out) {
  float hl[NLAYERS + 1];
  h_sequence(hl);
  const float c = hl[NLAYERS];
  v4f v = {c, c, c, c};
  const size_t base = (size_t)blockIdx.x * 1024 + threadIdx.x;
#pragma unroll
  for (int k = 0; k < 4; ++k)
    __builtin_nontemporal_store(v, &out[base + (size_t)k * 256]);
}

// ---- kernel 3: single-pass slab producer ----
// grid = (8, NSUB); each block owns a 32KB slice (2048 float4) of one sub-block.
__global__ __launch_bounds__(256) void slab_kernel(const v4f* __restrict__ in,
                                                   v4f* __restrict__ out,
                                                   const int* __restrict__ table) {
  __shared__ v4f buf[2048];  // 32 KB staging tile
  const int sb = blockIdx.y;
  const int mode = table[sb];  // uniform -> scalar load + scalar branch
  const size_t base =
      (size_t)sb * SUB_V4 + (size_t)blockIdx.x * 2048 + threadIdx.x;

  if (mode != -1) {
    const float c = __int_as_float(mode);
    v4f v = {c, c, c, c};
#pragma unroll
    for (int k = 0; k < 8; ++k)
      __builtin_nontemporal_store(v, &out[base + (size_t)k * 256]);
  } else {
#if defined(__HIP_DEVICE_COMPILE__)
    // 8 outstanding async B128 loads into LDS, fence, 8 async B128 stores out.
#pragma unroll
    for (int k = 0; k < 8; ++k)
      async_ld_b128(&buf[k * 256 + threadIdx.x], &in[base + (size_t)k * 256]);
    wait_async0();
#pragma unroll
    for (int k = 0; k < 8; ++k)
      async_st_b128(&out[base + (size_t)k * 256], &buf[k * 256 + threadIdx.x]);
    wait_async0();
#else
    for (int k = 0; k < 8; ++k)
      out[base + (size_t)k * 256] = in[base + (size_t)k * 256];
#endif
  }
}

extern "C" void kernel_launch(void* const* d_in, const int* in_sizes, int n_in,
                              void* d_out, int out_size, void* d_ws, size_t ws_size,
                              hipStream_t stream) {
  (void)in_sizes; (void)n_in; (void)out_size; (void)ws_size;
  // inputs: [0]=seq_lens (unused by reference), [1]=attn_block_ids, [2]=attn_page_slab
  const int* ids      = (const int*)d_in[1];
  const v4f* slab_in  = (const v4f*)d_in[2];
  float*     out      = (float*)d_out;
  v4f*       h_out    = (v4f*)out;                 // first H_ELEMS floats
  v4f*       slab_out = (v4f*)(out + H_ELEMS);     // then the slab
  int*       table    = (int*)d_ws;                // 1536 ints of scratch

  build_table<<<1, 512, 0, stream>>>(ids, table);
  fill_h_kernel<<<H_ELEMS / 4 / 1024, 256, 0, stream>>>(h_out);
  slab_kernel<<<dim3(8, NSUB), 256, 0, stream>>>(slab_in, slab_out, table);
}